// DenseGCN_7378753815022
// MI455X (gfx1250) — compile-verified
//
#include <hip/hip_runtime.h>

#define N_NODES 32768
#define KNN 16
#define GR 32
#define HST 160   // h row stride = 5*GR (dense-concat layout, concat is free)
#define UVST 64   // uv row stride: [u(32) | v(32)]

typedef float v2f __attribute__((ext_vector_type(2)));
typedef float v8f __attribute__((ext_vector_type(8)));

// h[:, 0:32] = x @ lin_x_w.T + lin_x_b      (K=3, trivial VALU kernel)
__global__ __launch_bounds__(256) void h0_kernel(const float* __restrict__ x,
                                                 const float* __restrict__ w,
                                                 const float* __restrict__ b,
                                                 float* __restrict__ h) {
  int t = blockIdx.x * 256 + threadIdx.x;
  int n = t >> 5, c = t & 31;
  const float* xr = x + n * 3;
  const float* wr = w + c * 3;
  h[(size_t)n * HST + c] =
      fmaf(xr[0], wr[0], fmaf(xr[1], wr[1], fmaf(xr[2], wr[2], b[c])));
}

// uv[:,0:32] = h[:,:C] @ (W1-W2)^T + b   (u part)
// uv[:,32:64] = h[:,:C] @ W2^T           (v part)
// One wave per 16-node tile; 4 x 16-channel output tiles; f32 WMMA 16x16x4.
template <int C>
__global__ __launch_bounds__(128) void edge_gemm(const float* __restrict__ h,
                                                 const float* __restrict__ w,    // (32, 2C)
                                                 const float* __restrict__ bias, // (32,)
                                                 float* __restrict__ uv) {
  // Build Wp = [[W1 - W2], [W2]]  (64 x C, row-major) directly in LDS.
  __shared__ float sW[64 * C];
  for (int idx = threadIdx.x; idx < 64 * C; idx += 128) {  // 64*C % 128 == 0: no divergence
    int c = idx / C, k = idx - c * C;
    float v;
    if (c < 32) v = w[c * 2 * C + k] - w[c * 2 * C + C + k];
    else        v = w[(c - 32) * 2 * C + C + k];
    sW[idx] = v;
  }
  __syncthreads();

  const int wave = threadIdx.x >> 5;
  const int lane = threadIdx.x & 31;
  const int half = lane >> 4;   // selects K pair {0,1} vs {2,3} per ISA A/B fragment layout
  const int l16  = lane & 15;
  const int node0 = (blockIdx.x * 4 + wave) * 16;

  v8f acc0 = {}, acc1 = {}, acc2 = {}, acc3 = {};
  const float* arow = h + (size_t)(node0 + l16) * HST + half * 2;  // A: M=l16 row, stream K
#pragma unroll
  for (int kk = 0; kk < C; kk += 4) {
    v2f a; a.x = arow[kk]; a.y = arow[kk + 1];
    const float* bp = sW + (size_t)l16 * C + kk + half * 2;        // B[k][n] = Wp[n][k]
    v2f b0; b0.x = bp[0];           b0.y = bp[1];
    v2f b1; b1.x = bp[16 * C + 0];  b1.y = bp[16 * C + 1];
    v2f b2; b2.x = bp[32 * C + 0];  b2.y = bp[32 * C + 1];
    v2f b3; b3.x = bp[48 * C + 0];  b3.y = bp[48 * C + 1];
    acc0 = __builtin_amdgcn_wmma_f32_16x16x4_f32(false, a, false, b0, (short)0, acc0, false, false);
    acc1 = __builtin_amdgcn_wmma_f32_16x16x4_f32(false, a, false, b1, (short)0, acc1, false, false);
    acc2 = __builtin_amdgcn_wmma_f32_16x16x4_f32(false, a, false, b2, (short)0, acc2, false, false);
    acc3 = __builtin_amdgcn_wmma_f32_16x16x4_f32(false, a, false, b3, (short)0, acc3, false, false);
  }

  // C/D layout: vgpr r, lane<16 -> (M=r, N=lane); lane>=16 -> (M=r+8, N=lane-16)
  float bv0 = bias[l16];
  float bv1 = bias[16 + l16];
#pragma unroll
  for (int r = 0; r < 8; ++r) {
    int row = r + half * 8;
    float* orow = uv + (size_t)(node0 + row) * UVST;
    orow[l16]      = acc0[r] + bv0;   // u channels 0..15  (+bias)
    orow[16 + l16] = acc1[r] + bv1;   // u channels 16..31 (+bias)
    orow[32 + l16] = acc2[r];         // v channels 0..15
    orow[48 + l16] = acc3[r];         // v channels 16..31
  }
}

// agg[n] = u[n] + max_k v[src[n + k*N]];  h[n, colOff:colOff+32] = agg[n]
// One wave per node; lane = channel. Neighbor indices broadcast via readlane
// so every v-row load is a uniform-base, fully coalesced 128B access.
__global__ __launch_bounds__(256) void gather_max_kernel(const int* __restrict__ src,
                                                         const float* __restrict__ uv,
                                                         float* __restrict__ h,
                                                         int colOff) {
  int wv = threadIdx.x >> 5, lane = threadIdx.x & 31;
  int node = blockIdx.x * 8 + wv;
  int jv = (lane < 16) ? src[node + lane * N_NODES] : 0;
  float m = -3.402823466e38f;
#pragma unroll
  for (int k = 0; k < 16; ++k) {
    int j = __builtin_amdgcn_readlane(jv, k);
    m = fmaxf(m, uv[(size_t)j * UVST + 32 + lane]);
  }
  float u = uv[(size_t)node * UVST + lane];
  h[(size_t)node * HST + colOff + lane] = u + m;
}

// out[n,c] = max over 5 dense-concat blocks of h[n, c + 32*b]
__global__ __launch_bounds__(256) void final_max_kernel(const float* __restrict__ h,
                                                        float* __restrict__ out) {
  int t = blockIdx.x * 256 + threadIdx.x;
  int n = t >> 5, c = t & 31;
  const float* hr = h + (size_t)n * HST + c;
  float m = hr[0];
#pragma unroll
  for (int bb = 1; bb < 5; ++bb) m = fmaxf(m, hr[bb * GR]);
  out[t] = m;
}

extern "C" void kernel_launch(void* const* d_in, const int* in_sizes, int n_in,
                              void* d_out, int out_size, void* d_ws, size_t ws_size,
                              hipStream_t stream) {
  const float* x   = (const float*)d_in[0];
  const int*   src = (const int*)d_in[1];  // edge_index row 0 (first E ints)
  const float* lxw = (const float*)d_in[2];
  const float* lxb = (const float*)d_in[3];
  const float* W[4] = {(const float*)d_in[4], (const float*)d_in[6],
                       (const float*)d_in[8], (const float*)d_in[10]};
  const float* B[4] = {(const float*)d_in[5], (const float*)d_in[7],
                       (const float*)d_in[9], (const float*)d_in[11]};

  float* h  = (float*)d_ws;                       // (N, 160)
  float* uv = h + (size_t)N_NODES * HST;          // (N, 64)

  h0_kernel<<<N_NODES * GR / 256, 256, 0, stream>>>(x, lxw, lxb, h);

  for (int i = 0; i < 4; ++i) {
    switch (i) {
      case 0: edge_gemm<32> <<<N_NODES / 64, 128, 0, stream>>>(h, W[0], B[0], uv); break;
      case 1: edge_gemm<64> <<<N_NODES / 64, 128, 0, stream>>>(h, W[1], B[1], uv); break;
      case 2: edge_gemm<96> <<<N_NODES / 64, 128, 0, stream>>>(h, W[2], B[2], uv); break;
      case 3: edge_gemm<128><<<N_NODES / 64, 128, 0, stream>>>(h, W[3], B[3], uv); break;
    }
    gather_max_kernel<<<N_NODES / 8, 256, 0, stream>>>(src, uv, h, GR * (i + 1));
  }

  final_max_kernel<<<N_NODES * GR / 256, 256, 0, stream>>>(h, (float*)d_out);
}